// GMMSynth_15616501088979
// MI455X (gfx1250) — compile-verified
//
#include <hip/hip_runtime.h>

// ---------------------------------------------------------------------------
// GMM synth: LUT remap + per-class masked noise field + per-class separable
// 3D Gaussian blur (radius 20, zero-padded) + class sum + final 3D blur.
// Conv-as-GEMM on the fp32 WMMA pipe: each 16x16 output tile of a 41-tap 1D
// conv is a 16x56 x 56x16 matmul (14 chained V_WMMA_F32_16X16X4_F32).
// LDS lines carry zero-filled radius-20 halos (unconditional ds_load_b64 A
// fragments); Toeplitz B fragments live in registers; generic-pass fill uses
// GLOBAL_LOAD_ASYNC_TO_LDS (ASYNCcnt) when the toolchain provides it.
// ---------------------------------------------------------------------------

typedef __attribute__((ext_vector_type(2))) float v2f;
typedef __attribute__((ext_vector_type(8))) float v8f;

#define DD 128
#define HH 160
#define WW 128
#define NVOX (DD * HH * WW)     // 2,621,440
#define RAD  20
#define KLEN 41
#define NCLS 20
#define SWPAD 72                // padded taps: [16 zeros][41 taps][15 zeros]
#define KCH 14                  // K = 56 = 14 chunks of 4

#if __has_builtin(__builtin_amdgcn_global_load_async_to_lds_b32)
#define HAS_ASYNC_LDS 1
typedef __attribute__((address_space(1))) int* as1_i32p;   // global
typedef __attribute__((address_space(3))) int* as3_i32p;   // LDS
#else
#define HAS_ASYNC_LDS 0
#endif

__device__ __forceinline__ void wait_async_all() {
#if HAS_ASYNC_LDS
#if __has_builtin(__builtin_amdgcn_s_wait_asynccnt)
    __builtin_amdgcn_s_wait_asynccnt(0);
#else
    asm volatile("s_wait_asynccnt 0x0" ::: "memory");
#endif
#endif
}

__constant__ int c_lut[36] = {
    0,1,2,3,4,5,6,7,8,9,10,11,12,13,14,15,16,17,18,19,
    1,2,3,4,5,6,7,8,9,10,14,15,16,17,18,19};

// ---- one-shot: remap labels to uint8 (read 20x later, 1B/voxel) ------------
__global__ void __launch_bounds__(256)
GMM_remap_kernel(const int* __restrict__ x, unsigned char* __restrict__ xr) {
    int i = (blockIdx.x * 256 + threadIdx.x) * 4;
    int4 v = *reinterpret_cast<const int4*>(x + i);
    uchar4 o;
    o.x = (unsigned char)c_lut[v.x];
    o.y = (unsigned char)c_lut[v.y];
    o.z = (unsigned char)c_lut[v.z];
    o.w = (unsigned char)c_lut[v.w];
    *reinterpret_cast<uchar4*>(xr + i) = o;
}

// ---- normalized truncated Gaussian taps for 20 class sigmas + final sigma --
__global__ void GMM_weights_kernel(const float* __restrict__ gsig,
                                   const float* __restrict__ fsig,
                                   float* __restrict__ wts) {
    int t = threadIdx.x;
    if (t <= NCLS) {
        float sig = (t < NCLS) ? gsig[t] : fsig[0];
        float tmp[KLEN];
        float s = 0.0f;
        #pragma unroll
        for (int k = 0; k < KLEN; ++k) {
            float d = (float)(k - RAD) / sig;
            float w = __expf(-0.5f * d * d);
            tmp[k] = w; s += w;
        }
        float inv = 1.0f / s;
        #pragma unroll
        for (int k = 0; k < KLEN; ++k) wts[t * KLEN + k] = tmp[k] * inv;
    }
}

// ---- pass 1 (per class): build field on the fly, conv along W via WMMA -----
#define LSF (WW + 2 * RAD + 2)   // 170: even (b64 align), 42 mod 64 -> conflict-free
__global__ void __launch_bounds__(256)
GMM_pass_w_fused(const unsigned char* __restrict__ xr,
                 const float* __restrict__ noise,
                 const float* __restrict__ mus, const float* __restrict__ stds,
                 const float* __restrict__ wts, float* __restrict__ out, int cls) {
    __shared__ float sl[16 * LSF];
    __shared__ float swp[SWPAD];
    const int tid = threadIdx.x;
    if (tid < SWPAD) {
        int t = tid - 16;
        swp[tid] = (t >= 0 && t < KLEN) ? wts[cls * KLEN + t] : 0.0f;
    }

    const int g0 = blockIdx.x * 16;               // first line (d*H + h)
    const float mu = mus[cls], sd = stds[cls];
    const float* np = noise + (size_t)cls * NVOX;

    // prefetch next block's noise stream (global_prefetch_b8)
    if (g0 + 16 < DD * HH) {
        const char* pf = (const char*)(np + (size_t)(g0 + 16) * WW);
        __builtin_prefetch(pf + tid * 32, 0, 1);
    }

    // fill 16 lines incl. zero halos; contiguous global reads per 16-lane group
    {
        const int line = tid >> 4;                // 16 threads per line
        const int v0 = (g0 + line) * WW;
        float* dst = &sl[line * LSF];
        for (int p = tid & 15; p < LSF; p += 16) {
            int e = p - RAD;
            float f = 0.0f;
            if (e >= 0 && e < WW) {
                int v = v0 + e;
                f = (xr[v] == (unsigned char)cls) ? (mu + sd * np[v]) : 0.0f;
            }
            dst[p] = f;
        }
    }
    __syncthreads();

    const int wave = tid >> 5, lane = tid & 31;
    const int m  = lane & 15;              // A row / D col index
    const int kk = (lane >> 4) << 1;       // K sub-offset for A/B fragments
    const int n  = lane & 15;              // output column

    v2f Bf[KCH];                           // Toeplitz B fragments in registers
    #pragma unroll
    for (int c = 0; c < KCH; ++c) {
        int t0 = 16 + c * 4 + kk - n;
        Bf[c] = (v2f){swp[t0], swp[t0 + 1]};
    }

    for (int tile = wave; tile < (WW >> 4); tile += 8) {
        const float* src = &sl[m * LSF + tile * 16];    // padded: idx>=0 always
        v2f Af[KCH];
        #pragma unroll
        for (int c = 0; c < KCH; ++c)                   // aligned ds_load_b64
            Af[c] = *reinterpret_cast<const v2f*>(&src[c * 4 + kk]);
        v8f acc = {0.f,0.f,0.f,0.f,0.f,0.f,0.f,0.f};
        #pragma unroll
        for (int c = 0; c < KCH; ++c)
            acc = __builtin_amdgcn_wmma_f32_16x16x4_f32(
                false, Af[c], false, Bf[c], (short)0, acc, false, false);
        #pragma unroll
        for (int r = 0; r < 8; ++r) {
            int M = (lane < 16) ? r : r + 8;
            out[(size_t)(g0 + M) * WW + tile * 16 + n] = acc[r];
        }
    }
}

// ---- generic 1D conv pass along an arbitrary axis (WMMA core) --------------
// line l base = (l/inner)*sOuter + (l%inner)*sInner ; element e at +e*eStride
// REQUIRES: inner % 16 == 0 (block of 16 lines never crosses an inner segment)
#define LSGMAX (HH + 2 * RAD + 2)       // 202 floats max padded stride
__global__ void __launch_bounds__(256)
GMM_pass_conv(const float* __restrict__ in, float* __restrict__ out,
              const float* __restrict__ wts, int widx, int L,
              int inner, long sOuter, long sInner, long eStride, int accum) {
    __shared__ float sl[16 * LSGMAX];
    __shared__ float swp[SWPAD];
    const int tid = threadIdx.x;
    if (tid < SWPAD) {
        int t = tid - 16;
        swp[tid] = (t >= 0 && t < KLEN) ? wts[widx * KLEN + t] : 0.0f;
    }

    const int ls = L + 2 * RAD + 2;       // even; 170 or 202 -> conflict-free
    const int g0 = blockIdx.x * 16;
    // one scalar div for the whole block (inner % 16 == 0, g0 % 16 == 0)
    const int  q  = g0 / inner;
    const int  rm = g0 - q * inner;
    const long baseBlock = (long)q * sOuter + (long)rm * sInner;

    // fill: pick thread->(line,p) mapping so consecutive lanes coalesce
    {
        int line, p0;
        if (eStride == 1) { line = tid >> 4; p0 = tid & 15; }   // e-fastest
        else              { line = tid & 15; p0 = tid >> 4; }   // line-fastest
        const long base = baseBlock + (long)line * sInner;
        const float* srcg = in + base;
        float* dst = &sl[line * ls];
        for (int p = p0; p < ls; p += 16) {
            int e = p - RAD;
            if (e >= 0 && e < L) {
#if HAS_ASYNC_LDS
                __builtin_amdgcn_global_load_async_to_lds_b32(
                    (as1_i32p)(unsigned long long)(srcg + (long)e * eStride),
                    (as3_i32p)(unsigned long long)(dst + p), 0, 0);
#else
                dst[p] = srcg[(long)e * eStride];
#endif
            } else {
                dst[p] = 0.0f;
            }
        }
    }
    wait_async_all();
    __syncthreads();

    const int wave = tid >> 5, lane = tid & 31;
    const int m  = lane & 15;
    const int kk = (lane >> 4) << 1;
    const int n  = lane & 15;

    v2f Bf[KCH];
    #pragma unroll
    for (int c = 0; c < KCH; ++c) {
        int t0 = 16 + c * 4 + kk - n;
        Bf[c] = (v2f){swp[t0], swp[t0 + 1]};
    }

    const int nT = L >> 4;
    for (int tile = wave; tile < nT; tile += 8) {
        const float* src = &sl[m * ls + tile * 16];
        v2f Af[KCH];
        #pragma unroll
        for (int c = 0; c < KCH; ++c)
            Af[c] = *reinterpret_cast<const v2f*>(&src[c * 4 + kk]);
        v8f acc = {0.f,0.f,0.f,0.f,0.f,0.f,0.f,0.f};
        #pragma unroll
        for (int c = 0; c < KCH; ++c)
            acc = __builtin_amdgcn_wmma_f32_16x16x4_f32(
                false, Af[c], false, Bf[c], (short)0, acc, false, false);
        const long obase = baseBlock + (long)(tile * 16 + n) * eStride;
        #pragma unroll
        for (int r = 0; r < 8; ++r) {
            int M = (lane < 16) ? r : r + 8;
            long oi = obase + (long)M * sInner;
            if (accum) out[oi] += acc[r];
            else       out[oi]  = acc[r];
        }
    }
}

extern "C" void kernel_launch(void* const* d_in, const int* in_sizes, int n_in,
                              void* d_out, int out_size, void* d_ws, size_t ws_size,
                              hipStream_t stream) {
    const int*   x     = (const int*)  d_in[0];
    const float* noise = (const float*)d_in[1];
    const float* mus   = (const float*)d_in[2];
    const float* stds  = (const float*)d_in[3];
    const float* gsig  = (const float*)d_in[4];
    const float* fsig  = (const float*)d_in[5];
    float* outp = (float*)d_out;

    char*  ws   = (char*)d_ws;
    float* wts  = (float*)ws;                                    // 21*41 floats
    float* tmpA = (float*)(ws + 4096);                           // NVOX f32
    float* tmpB = (float*)(ws + 4096 + (size_t)NVOX * 4);        // NVOX f32
    float* sum  = (float*)(ws + 4096 + (size_t)NVOX * 8);        // NVOX f32
    unsigned char* xr = (unsigned char*)(ws + 4096 + (size_t)NVOX * 12); // NVOX u8

    GMM_weights_kernel<<<1, 64, 0, stream>>>(gsig, fsig, wts);
    GMM_remap_kernel<<<NVOX / 1024, 256, 0, stream>>>(x, xr);

    for (int cls = 0; cls < NCLS; ++cls) {
        // W axis, field fused in (lines = d*H+h)
        GMM_pass_w_fused<<<DD * HH / 16, 256, 0, stream>>>(
            xr, noise, mus, stds, wts, tmpA, cls);
        // H axis: lines (d,w): base = d*H*W + w, elem stride W, L=H
        GMM_pass_conv<<<DD * WW / 16, 256, 0, stream>>>(
            tmpA, tmpB, wts, cls, HH, WW, (long)HH * WW, 1L, (long)WW, 0);
        // D axis: lines (h,w): base = h*W + w, elem stride H*W, L=D; accumulate
        GMM_pass_conv<<<HH * WW / 16, 256, 0, stream>>>(
            tmpB, sum, wts, cls, DD, HH * WW, 0L, 1L, (long)HH * WW, cls ? 1 : 0);
    }

    // final smooth with sigma index 20: W, H, D
    GMM_pass_conv<<<DD * HH / 16, 256, 0, stream>>>(
        sum, tmpA, wts, NCLS, WW, DD * HH, 0L, (long)WW, 1L, 0);
    GMM_pass_conv<<<DD * WW / 16, 256, 0, stream>>>(
        tmpA, tmpB, wts, NCLS, HH, WW, (long)HH * WW, 1L, (long)WW, 0);
    GMM_pass_conv<<<HH * WW / 16, 256, 0, stream>>>(
        tmpB, outp, wts, NCLS, DD, HH * WW, 0L, 1L, (long)HH * WW, 0);
}